// DNN_sym_10101763080772
// MI455X (gfx1250) — compile-verified
//
#include <hip/hip_runtime.h>
#include <hip/hip_bf16.h>

typedef __attribute__((ext_vector_type(2))) float v2f;
typedef __attribute__((ext_vector_type(4))) float v4f;
typedef __attribute__((ext_vector_type(8))) float v8f;

#define HL 8192

// ---------------------------------------------------------------------------
// Kernel 1: per-atom routed 3x3 embed + outer-product reduction -> d[9]
//   g_i = (atom_list[i]==1 ? W1 : W12) @ x_i + b,  d[a*3+c] = sum_i g_i[a]*x_i[c]
// ---------------------------------------------------------------------------
__global__ void __launch_bounds__(128)
k_embed(const float* __restrict__ x,
        const float* __restrict__ W1,  const float* __restrict__ b1,
        const float* __restrict__ W12, const float* __restrict__ b12,
        const int*   __restrict__ atom_list,
        float* __restrict__ d_vec) {
  __shared__ float red[9 * 128];
  const int t = threadIdx.x;                // 0..127 == N_ATOMS
  const float x0 = x[t * 3 + 0];
  const float x1 = x[t * 3 + 1];
  const float x2 = x[t * 3 + 2];
  const bool sel = (atom_list[t] == 1);
  const float* W = sel ? W1 : W12;
  const float* b = sel ? b1 : b12;
  float g[3];
#pragma unroll
  for (int a = 0; a < 3; ++a)
    g[a] = b[a] + W[a * 3 + 0] * x0 + W[a * 3 + 1] * x1 + W[a * 3 + 2] * x2;
  const float xs[3] = {x0, x1, x2};
#pragma unroll
  for (int a = 0; a < 3; ++a)
#pragma unroll
    for (int c = 0; c < 3; ++c)
      red[(a * 3 + c) * 128 + t] = g[a] * xs[c];
  __syncthreads();
  for (int s = 64; s > 0; s >>= 1) {
    if (t < s) {
#pragma unroll
      for (int p = 0; p < 9; ++p) red[p * 128 + t] += red[p * 128 + t + s];
    }
    __syncthreads();
  }
  if (t < 9) d_vec[t] = red[t * 128];
}

// ---------------------------------------------------------------------------
// Kernel 2: h0 = Wl0 @ d + bl0   (8192 x 9 -- tiny, one thread per row)
// ---------------------------------------------------------------------------
__global__ void __launch_bounds__(256)
k_layer0(const float* __restrict__ Wl0, const float* __restrict__ bl0,
         const float* __restrict__ d_vec, float* __restrict__ h) {
  const int o = blockIdx.x * blockDim.x + threadIdx.x;
  float dv[9];
#pragma unroll
  for (int j = 0; j < 9; ++j) dv[j] = d_vec[j];
  const float* w = Wl0 + (long long)o * 9;
  float acc = bl0[o];
#pragma unroll
  for (int j = 0; j < 9; ++j) acc += w[j] * dv[j];
  h[o] = acc;
}

// ---------------------------------------------------------------------------
// Kernel 3: 8192x8192 GEMV slice via V_WMMA_F32_16X16X4_F32.
// Grid = 64 row-blocks x 2^lgks k-slices; each wave owns 16 rows x klen.
// Partials go to part[kslice][row]; fixed-order reduce in k_reduce keeps
// results deterministic. Two accumulators hide the XDL D->C chain.
// A-fragment (ISA 16x4 f32): lanes 0-15 V0/V1 = K0/K1, lanes 16-31 = K2/K3.
// B broadcasts h[k] to all 16 columns, so every D column holds the result.
// ---------------------------------------------------------------------------
__global__ void __launch_bounds__(256)
k_gemv_wmma(const float* __restrict__ W,
            const float* __restrict__ hin,
            float* __restrict__ part,       // [2^lgks][HL]
            int klen, int lgks) {
  __shared__ float lh[HL];                  // only klen floats used
  __builtin_assume(klen >= 1024);
  __builtin_assume((klen & 1023) == 0);

  const int ksl = blockIdx.x & ((1 << lgks) - 1);
  const int rb  = blockIdx.x >> lgks;       // 0..63 (128 rows each)
  const int k0  = ksl * klen;

  {                                         // stage this block's h slice
    const v4f* src = (const v4f*)(hin + k0);
    v4f* dst = (v4f*)lh;
    for (int i = threadIdx.x; i < (klen >> 2); i += blockDim.x) dst[i] = src[i];
  }
  __syncthreads();

  const int lane = threadIdx.x & 31;
  const int half = lane >> 4;               // 0: K+{0,1}, 1: K+{2,3}
  const int kb   = half * 2;
  const int rows0 = rb * 128 + (threadIdx.x >> 5) * 16;
  const long long row = (long long)(rows0 + (lane & 15));
  const float* wr = W + row * HL + k0;

  v8f c0 = {0.f, 0.f, 0.f, 0.f, 0.f, 0.f, 0.f, 0.f};
  v8f c1 = {0.f, 0.f, 0.f, 0.f, 0.f, 0.f, 0.f, 0.f};
  for (int kk = 0; kk < klen; kk += 16) {
    if (kk + 128 < klen) __builtin_prefetch(wr + kk + 128, 0, 0);
    v2f a0 = *(const v2f*)(wr + kk + 0  + kb);
    v2f a1 = *(const v2f*)(wr + kk + 4  + kb);
    v2f a2 = *(const v2f*)(wr + kk + 8  + kb);
    v2f a3 = *(const v2f*)(wr + kk + 12 + kb);
    v2f bb0 = *(const v2f*)(lh + kk + 0  + kb);
    v2f bb1 = *(const v2f*)(lh + kk + 4  + kb);
    v2f bb2 = *(const v2f*)(lh + kk + 8  + kb);
    v2f bb3 = *(const v2f*)(lh + kk + 12 + kb);
    c0 = __builtin_amdgcn_wmma_f32_16x16x4_f32(false, a0, false, bb0, (short)0, c0, false, false);
    c1 = __builtin_amdgcn_wmma_f32_16x16x4_f32(false, a1, false, bb1, (short)0, c1, false, false);
    c0 = __builtin_amdgcn_wmma_f32_16x16x4_f32(false, a2, false, bb2, (short)0, c0, false, false);
    c1 = __builtin_amdgcn_wmma_f32_16x16x4_f32(false, a3, false, bb3, (short)0, c1, false, false);
  }
  c0 += c1;

  // D layout: lane 0 V0..7 = rows m..m+7 ; lane 16 V0..7 = rows m+8..m+15
  if ((lane & 15) == 0) {
    const int base = rows0 + half * 8;
    float* po = part + ((long long)ksl << 13) + base;   // ksl*HL
#pragma unroll
    for (int v = 0; v < 8; ++v) po[v] = c0[v];
  }
}

// ---------------------------------------------------------------------------
// Kernel 4: hout[o] = bias[o] + sum_s part[s][o]   (fixed-order, deterministic)
// ---------------------------------------------------------------------------
__global__ void __launch_bounds__(256)
k_reduce(const float* __restrict__ part, const float* __restrict__ bias,
         float* __restrict__ hout, int kslices) {
  const int o = blockIdx.x * blockDim.x + threadIdx.x;
  float acc = bias[o];
  for (int s = 0; s < kslices; ++s) acc += part[((long long)s << 13) + o];
  hout[o] = acc;
}

// ---------------------------------------------------------------------------
// Kernel 5: out[3] = Wo @ h + bo   (3 x 8192 reduction, one block)
// ---------------------------------------------------------------------------
__global__ void __launch_bounds__(256)
k_out(const float* __restrict__ Wo, const float* __restrict__ bo,
      const float* __restrict__ h, float* __restrict__ out) {
  __shared__ float red[3 * 256];
  const int t = threadIdx.x;
  float a0 = 0.f, a1 = 0.f, a2 = 0.f;
  for (int k = t; k < HL; k += 256) {
    const float hv = h[k];
    a0 += Wo[0 * HL + k] * hv;
    a1 += Wo[1 * HL + k] * hv;
    a2 += Wo[2 * HL + k] * hv;
  }
  red[0 * 256 + t] = a0;
  red[1 * 256 + t] = a1;
  red[2 * 256 + t] = a2;
  __syncthreads();
  for (int s = 128; s > 0; s >>= 1) {
    if (t < s) {
      red[0 * 256 + t] += red[0 * 256 + t + s];
      red[1 * 256 + t] += red[1 * 256 + t + s];
      red[2 * 256 + t] += red[2 * 256 + t + s];
    }
    __syncthreads();
  }
  if (t < 3) out[t] = red[t * 256] + bo[t];
}

// ---------------------------------------------------------------------------
extern "C" void kernel_launch(void* const* d_in, const int* in_sizes, int n_in,
                              void* d_out, int out_size, void* d_ws, size_t ws_size,
                              hipStream_t stream) {
  const float* x   = (const float*)d_in[0];
  const float* W1  = (const float*)d_in[1];
  const float* b1  = (const float*)d_in[2];
  const float* W12 = (const float*)d_in[3];
  const float* b12 = (const float*)d_in[4];
  const float* Wl0 = (const float*)d_in[5];
  const float* bl0 = (const float*)d_in[6];
  const float* Wl1 = (const float*)d_in[7];
  const float* bl1 = (const float*)d_in[8];
  const float* Wl2 = (const float*)d_in[9];
  const float* bl2 = (const float*)d_in[10];
  const float* Wo  = (const float*)d_in[11];
  const float* bo  = (const float*)d_in[12];
  const int* atom_list = (const int*)d_in[13];

  // k-split factor: as many concurrent waves as the workspace allows
  // (deterministic: depends only on the fixed ws_size).
  int lgks = 3;                              // kslices = 8 preferred
  while (lgks > 0 &&
         (size_t)(64 + 2 * HL + ((size_t)HL << lgks)) * sizeof(float) > ws_size)
    --lgks;
  const int kslices = 1 << lgks;
  const int klen = HL >> lgks;

  float* ws   = (float*)d_ws;
  float* dvec = ws;                 // 9 used (64 reserved)
  float* hA   = ws + 64;            // 8192
  float* hB   = ws + 64 + HL;       // 8192
  float* part = ws + 64 + 2 * HL;   // kslices * 8192
  float* out  = (float*)d_out;

  k_embed    <<<1, 128, 0, stream>>>(x, W1, b1, W12, b12, atom_list, dvec);
  k_layer0   <<<HL / 256, 256, 0, stream>>>(Wl0, bl0, dvec, hA);

  k_gemv_wmma<<<64 * kslices, 256, 0, stream>>>(Wl1, hA, part, klen, lgks);
  k_reduce   <<<HL / 256, 256, 0, stream>>>(part, bl1, hB, kslices);

  k_gemv_wmma<<<64 * kslices, 256, 0, stream>>>(Wl2, hB, part, klen, lgks);
  k_reduce   <<<HL / 256, 256, 0, stream>>>(part, bl2, hA, kslices);

  k_out      <<<1, 256, 0, stream>>>(Wo, bo, hA, out);
}